// SOMLayer_63728724738463
// MI455X (gfx1250) — compile-verified
//
#include <hip/hip_runtime.h>

typedef __attribute__((ext_vector_type(2))) float v2f;
typedef __attribute__((ext_vector_type(8))) float v8f;

#define N_SAMPLES 8192
#define LATENT    32
#define P_UNITS   1024
#define PRSTRIDE  96   // LDS floats per K-pair row (64 data + 32 pad): 96 = 32 mod 64 banks

// ---------------------------------------------------------------------------
// Kernel 1: row norms  xn2[n] = ||x_n||^2 , wp2[p] = ||w_p||^2
// ---------------------------------------------------------------------------
__global__ void __launch_bounds__(256)
som_norms_kernel(const float* __restrict__ X, const float* __restrict__ W,
                 float* __restrict__ xn2, float* __restrict__ wp2) {
  int t = blockIdx.x * blockDim.x + threadIdx.x;
  if (t < N_SAMPLES) {
    const float* row = X + (size_t)t * LATENT;
    float s = 0.f;
#pragma unroll
    for (int k = 0; k < LATENT; ++k) s += row[k] * row[k];
    xn2[t] = s;
  } else if (t < N_SAMPLES + P_UNITS) {
    int p = t - N_SAMPLES;
    const float* row = W + (size_t)p * LATENT;
    float s = 0.f;
#pragma unroll
    for (int k = 0; k < LATENT; ++k) s += row[k] * row[k];
    wp2[p] = s;
  }
}

// ---------------------------------------------------------------------------
// Kernel 2: dTi = K-pair-interleaved distance matrix:
//   dTi[(p>>1)*2N + 2n + (p&1)] = ||x_n||^2 + ||w_p||^2 - 2 (X W^T)[n][p]
// One wave = one 16x16 (n,p) tile via 8x v_wmma_f32_16x16x4_f32.
// ---------------------------------------------------------------------------
__global__ void __launch_bounds__(256)
som_dist_kernel(const float* __restrict__ X, const float* __restrict__ W,
                const float* __restrict__ xn2, const float* __restrict__ wp2,
                float* __restrict__ dTi) {
  const int lane = threadIdx.x & 31;
  const int wave = threadIdx.x >> 5;
  const int r    = lane & 15;
  const int half = lane >> 4;
  const int n0   = blockIdx.x * 16;
  const int p0   = (blockIdx.y * 8 + wave) * 16;

  const float* xrow = X + (size_t)(n0 + r) * LATENT + 2 * half;
  const float* wrow = W + (size_t)(p0 + r) * LATENT + 2 * half;

  v8f acc = {};
#pragma unroll
  for (int kc = 0; kc < 8; ++kc) {
    v2f a = *(const v2f*)(xrow + 4 * kc);
    v2f b = *(const v2f*)(wrow + 4 * kc);
    acc = __builtin_amdgcn_wmma_f32_16x16x4_f32(false, a, false, b,
                                                (short)0, acc, false, false);
  }

  const int p  = p0 + r;
  const float wq = wp2[p];
  float* dbase = dTi + (size_t)(p >> 1) * (2 * N_SAMPLES)
                     + (size_t)(n0 + 8 * half) * 2 + (p & 1);
#pragma unroll
  for (int i = 0; i < 8; ++i) {
    const int n = n0 + 8 * half + i;           // C layout: M = i + 8*half
    dbase[2 * i] = xn2[n] + wq - 2.0f * acc[i];
  }
}

// ---------------------------------------------------------------------------
// Kernel 3: E = h (PxP) @ dT (PxN); per-column argmin over q plus loss.
// Block (8 waves) owns a 32-column strip. The K-pair-interleaved strip
// (512 pair-rows x 64 floats, 128 KB) is staged into LDS with async
// global->LDS b128 copies; each B fragment is then ONE bank-conflict-free
// ds_load_b64. Waves sweep their 128-q range with 2x2 register-tiled WMMA.
// ---------------------------------------------------------------------------
__device__ __forceinline__ void upd_min(float v, int q, float& bv, int& bq) {
  if (v < bv || (v == bv && q < bq)) { bv = v; bq = q; }
}

__global__ void __launch_bounds__(256)
som_energy_argmin_kernel(const float* __restrict__ H, const float* __restrict__ dTi,
                         float* __restrict__ out) {
  extern __shared__ float sd[];                // 512 * PRSTRIDE floats (192 KB)
  __shared__ float lv[16][32];
  __shared__ int   li[16][32];

  const int t    = threadIdx.x;
  const int lane = t & 31;
  const int wave = t >> 5;
  const int r    = lane & 15;
  const int half = lane >> 4;
  const int n0   = blockIdx.x * 32;

  // ---- async cooperative copy: strip cols n0..n0+31 of all 512 pair-rows.
  // One pair-row slice = 64 consecutive floats (256 B) in global; 16 threads
  // per pair-row, each moving one contiguous 16 B chunk.
#pragma unroll 4
  for (int j = 0; j < 32; ++j) {
    const int idx = t + 256 * j;               // 0..8191
    const int pr  = idx >> 4;                  // pair-row 0..511
    const int c4  = (idx & 15) * 4;            // float offset within 64-float slice
    const unsigned lds = (unsigned)(unsigned long long)&sd[pr * PRSTRIDE + c4];
    const float* g = dTi + (size_t)pr * (2 * N_SAMPLES) + 2 * n0 + c4;
    asm volatile("global_load_async_to_lds_b128 %0, %1, off"
                 :: "v"(lds), "v"(g) : "memory");
  }
  asm volatile("s_wait_asynccnt 0x0" ::: "memory");
  __syncthreads();

  float bestv0 = 3.402823e38f, bestv1 = 3.402823e38f;
  int   bestq0 = 0,            bestq1 = 0;

  for (int qb = 0; qb < 4; ++qb) {
    const int q0 = wave * 128 + qb * 32;
    v8f acc00 = {}, acc01 = {}, acc10 = {}, acc11 = {};

    const float* hrow0 = H + (size_t)(q0 + r) * P_UNITS + 2 * half;   // q tile 0
    const float* hrow1 = hrow0 + (size_t)16 * P_UNITS;                // q tile 1

#pragma unroll 4
    for (int p4 = 0; p4 < P_UNITS; p4 += 4) {
      v2f a0 = *(const v2f*)(hrow0 + p4);
      v2f a1 = *(const v2f*)(hrow1 + p4);
      const int s0 = ((p4 >> 1) + half) * PRSTRIDE + 2 * r;  // K pair for this half
      v2f b0 = *(const v2f*)&sd[s0];                         // cols r   : ds_load_b64
      v2f b1 = *(const v2f*)&sd[s0 + 32];                    // cols 16+r: ds_load_b64
      acc00 = __builtin_amdgcn_wmma_f32_16x16x4_f32(false, a0, false, b0,
                                                    (short)0, acc00, false, false);
      acc01 = __builtin_amdgcn_wmma_f32_16x16x4_f32(false, a0, false, b1,
                                                    (short)0, acc01, false, false);
      acc10 = __builtin_amdgcn_wmma_f32_16x16x4_f32(false, a1, false, b0,
                                                    (short)0, acc10, false, false);
      acc11 = __builtin_amdgcn_wmma_f32_16x16x4_f32(false, a1, false, b1,
                                                    (short)0, acc11, false, false);
    }

#pragma unroll
    for (int i = 0; i < 8; ++i) {
      const int q = q0 + 8 * half + i;         // C layout: M = i + 8*half
      upd_min(acc00[i], q,      bestv0, bestq0);   // col = r       (n tile 0)
      upd_min(acc10[i], q + 16, bestv0, bestq0);
      upd_min(acc01[i], q,      bestv1, bestq1);   // col = 16 + r  (n tile 1)
      upd_min(acc11[i], q + 16, bestv1, bestq1);
    }
  }

  const int slot = wave * 2 + half;            // 16 candidate sets per column
  lv[slot][r]      = bestv0;  li[slot][r]      = bestq0;
  lv[slot][16 + r] = bestv1;  li[slot][16 + r] = bestq1;
  __syncthreads();

  if (t < 32) {
    const int c = t;
    float bv = lv[0][c]; int bq = li[0][c];
#pragma unroll
    for (int s = 1; s < 16; ++s) {
      const float v = lv[s][c]; const int q = li[s][c];
      if (v < bv || (v == bv && q < bq)) { bv = v; bq = q; }
    }
    const int n = n0 + c;
    out[n]             = (float)bq;            // bmus (return-order slot 0)
    out[N_SAMPLES + n] = 0.5f * bv;            // loss (return-order slot 1)
  }
}

// ---------------------------------------------------------------------------
extern "C" void kernel_launch(void* const* d_in, const int* in_sizes, int n_in,
                              void* d_out, int out_size, void* d_ws, size_t ws_size,
                              hipStream_t stream) {
  (void)in_sizes; (void)n_in; (void)out_size; (void)ws_size;
  const float* X = (const float*)d_in[0];   // (8192, 32)
  const float* W = (const float*)d_in[1];   // (1024, 32)
  const float* H = (const float*)d_in[2];   // (1024, 1024)
  float* out = (float*)d_out;               // [bmus(8192) | loss(8192)]

  float* dTi = (float*)d_ws;                           // interleaved (P/2, N, 2) = 32 MB
  float* xn2 = dTi + (size_t)P_UNITS * N_SAMPLES;      // 8192 floats
  float* wp2 = xn2 + N_SAMPLES;                        // 1024 floats

  som_norms_kernel<<<(N_SAMPLES + P_UNITS + 255) / 256, 256, 0, stream>>>(
      X, W, xn2, wp2);

  dim3 g2(N_SAMPLES / 16, P_UNITS / (16 * 8));         // (512, 8), 8 waves/block
  som_dist_kernel<<<g2, 256, 0, stream>>>(X, W, xn2, wp2, dTi);

  const size_t smem = (size_t)512 * PRSTRIDE * sizeof(float);  // 192 KB
  som_energy_argmin_kernel<<<N_SAMPLES / 32, 256, smem, stream>>>(H, dTi, out);
}